// SeparateHiddenPRADA_369367188156
// MI455X (gfx1250) — compile-verified
//
#include <hip/hip_runtime.h>
#include <hip/hip_bf16.h>
#include <cmath>

// CDNA5 / gfx1250: wave32, WMMA 16x16x4 F32 path.
typedef __attribute__((ext_vector_type(2))) float v2f;
typedef __attribute__((ext_vector_type(8))) float v8f;

__device__ __forceinline__ void atomAddF(float* p, float v) {
  // lowers to global_atomic_add_f32 on gfx1250 (no CAS loop)
  unsafeAtomicAdd(p, v);
}

// ---------------------------------------------------------------- elementwise
__global__ void k_fill(float* __restrict__ p, float v, int n) {
  int i = blockIdx.x * blockDim.x + threadIdx.x;
  if (i < n) p[i] = v;
}

__global__ void k_count_deg(float* __restrict__ deg, const int* __restrict__ col, int e) {
  int i = blockIdx.x * blockDim.x + threadIdx.x;
  if (i < e) atomAddF(&deg[col[i]], 1.0f);
}

__global__ void k_rsqrt(float* __restrict__ p, int n) {
  int i = blockIdx.x * blockDim.x + threadIdx.x;
  if (i < n) p[i] = rsqrtf(p[i]);
}

// out[i, colOff+j] = bias[j] + T[i,j] * dis[i]^2   (self-loop + bias, full init)
__global__ void k_init_out(float* __restrict__ out, const float* __restrict__ T,
                           const float* __restrict__ dis, const float* __restrict__ bias,
                           int nRows, int width, int ldOut, int colOff) {
  int idx = blockIdx.x * blockDim.x + threadIdx.x;
  if (idx >= nRows * width) return;
  int i = idx / width;
  int j = idx - i * width;
  float d = dis[i];
  out[(size_t)i * ldOut + colOff + j] = bias[j] + T[idx] * d * d;
}

__global__ void k_tanh_ip(float* __restrict__ h, int nRows, int width, int ldOut, int colOff) {
  int idx = blockIdx.x * blockDim.x + threadIdx.x;
  if (idx >= nRows * width) return;
  int i = idx / width;
  int j = idx - i * width;
  float* p = h + (size_t)i * ldOut + colOff + j;
  *p = tanhf(*p);
}

// z = noise * exp(0.5*logvar) + mean
__global__ void k_z(float* __restrict__ z, const float* __restrict__ mean,
                    const float* __restrict__ logvar, const float* __restrict__ noise, int n) {
  int i = blockIdx.x * blockDim.x + threadIdx.x;
  if (i < n) z[i] = noise[i] * expf(0.5f * logvar[i]) + mean[i];
}

// ------------------------------------------------------------------- scatter
// One thread per (edge, 4-float chunk): wave32 covers a 128-wide row of one
// edge. Coalesced float4 gather of T[row], f32 atomics into out[col] (L2).
__global__ void k_scatter(float* __restrict__ out, const float* __restrict__ T,
                          const float* __restrict__ dis,
                          const int* __restrict__ row, const int* __restrict__ col,
                          int e, int width, int ldOut, int colOff) {
  int chunks = width >> 2;
  long long idx = (long long)blockIdx.x * blockDim.x + threadIdx.x;
  long long total = (long long)e * chunks;
  if (idx >= total) return;
  int ei = (int)(idx / chunks);
  int c  = (int)(idx - (long long)ei * chunks);
  int r  = row[ei];
  int cl = col[ei];
  float w = dis[r] * dis[cl];
  const float4 v = *(const float4*)(T + (size_t)r * width + (size_t)c * 4);
  float* o = out + (size_t)cl * ldOut + colOff + (size_t)c * 4;
  atomAddF(o + 0, v.x * w);
  atomAddF(o + 1, v.y * w);
  atomAddF(o + 2, v.z * w);
  atomAddF(o + 3, v.w * w);
  // warm L2 for a later edge's source row (global_prefetch_b8 on gfx1250)
  int ep = ei + 64;
  if (ep < e) __builtin_prefetch(T + (size_t)row[ep] * width + (size_t)c * 4, 0, 1);
}

// ---------------------------------------------------------------- WMMA GEMM
// T[M,Hout] = X[M,K] @ W[K,Hout], fp32 via V_WMMA_F32_16X16X4_F32.
// One wave -> one 16-row stripe across ALL NT = Hout/16 column tiles:
// the A-fragment (X rows) is loaded once per k-step and reused for NT
// back-to-back WMMAs into NT register accumulators (NT<=8 -> 64 acc VGPRs).
// 4 waves per block over consecutive M tiles.
// A 16x4 layout: lanes 0-15 hold rows M, VGPR{0,1} = K{0,1}; lanes 16-31 = K{2,3}.
// B 4x16 layout: lanes hold cols N with the same K split.
// C/D: VGPR v -> row (v + 8*(lane>=16)), col = lane&15.
template <int NT>
__global__ void k_gemm_wmma(const float* __restrict__ X, const float* __restrict__ W,
                            float* __restrict__ T, int nRows, int K) {
  const int Hout = NT * 16;
  int wave = threadIdx.x >> 5;
  int lane = threadIdx.x & 31;
  int half = lane >> 4;      // 0: K offset 0/1, 1: K offset 2/3
  int l    = lane & 15;
  int tileM = blockIdx.x * 4 + wave;
  int m0 = tileM * 16;
  if (m0 >= nRows) return;           // wave-uniform: EXEC stays all-ones below

  int mA = m0 + l;
  if (mA >= nRows) mA = nRows - 1;   // clamp loads, stores are guarded
  const float* xrow = X + (size_t)mA * K;

  v8f acc[NT];
#pragma unroll
  for (int t = 0; t < NT; ++t) acc[t] = (v8f){};

  for (int k = 0; k < K; k += 4) {
    int ka = k + half * 2;
    v2f a;
    a.x = xrow[ka];
    a.y = xrow[ka + 1];
    const float* w0 = W + (size_t)ka * Hout + l;
    const float* w1 = W + (size_t)(ka + 1) * Hout + l;
#pragma unroll
    for (int t = 0; t < NT; ++t) {
      v2f b;
      b.x = w0[t * 16];
      b.y = w1[t * 16];
      acc[t] = __builtin_amdgcn_wmma_f32_16x16x4_f32(
          /*neg_a=*/false, a, /*neg_b=*/false, b,
          /*c_mod=*/(short)0, acc[t], /*reuse_a=*/false, /*reuse_b=*/false);
    }
  }
#pragma unroll
  for (int t = 0; t < NT; ++t) {
#pragma unroll
    for (int v = 0; v < 8; ++v) {
      int m = m0 + v + half * 8;
      if (m < nRows) T[(size_t)m * Hout + t * 16 + l] = acc[t][v];
    }
  }
}

// ------------------------------------------------------------------- launch
extern "C" void kernel_launch(void* const* d_in, const int* in_sizes, int n_in,
                              void* d_out, int out_size, void* d_ws, size_t ws_size,
                              hipStream_t stream) {
  (void)n_in; (void)out_size; (void)ws_size;
  const int F = 128, C = 64, H = 128, L = 32;

  const float* feature   = (const float*)d_in[0];
  const float* condition = (const float*)d_in[1];
  const float* noise     = (const float*)d_in[2];
  const float* W_f2h     = (const float*)d_in[3];
  const float* b_f2h     = (const float*)d_in[4];
  const float* W_c2h_e   = (const float*)d_in[5];
  const float* b_c2h_e   = (const float*)d_in[6];
  const float* W_mean    = (const float*)d_in[7];
  const float* b_mean    = (const float*)d_in[8];
  const float* W_logvar  = (const float*)d_in[9];
  const float* b_logvar  = (const float*)d_in[10];
  const float* W_z2h     = (const float*)d_in[11];
  const float* b_z2h     = (const float*)d_in[12];
  const float* W_c2h_d   = (const float*)d_in[13];
  const float* b_c2h_d   = (const float*)d_in[14];
  const float* W_out     = (const float*)d_in[15];
  const float* b_out     = (const float*)d_in[16];
  const int*   edges     = (const int*)d_in[17];

  const int n = in_sizes[0] / F;
  const int e = in_sizes[17] / 2;
  const int* rowIdx = edges;       // edge_index[0]
  const int* colIdx = edges + e;   // edge_index[1]

  // workspace layout (floats): dis[n] | T[n*128] | h[n*256] | hd[n*256]
  float* ws  = (float*)d_ws;
  float* dis = ws;
  float* T   = dis + n;
  float* h   = T + (size_t)n * 128;
  float* hd  = h + (size_t)n * 256;

  // d_out layout: z[n*32] | mean[n*32] | logvar[n*32] | out[n*128]
  float* zOut    = (float*)d_out;
  float* meanOut = zOut + (size_t)n * L;
  float* lvOut   = meanOut + (size_t)n * L;
  float* outOut  = lvOut + (size_t)n * L;

  const int TB = 256;

  // ---- degree / dis = (deg+1)^-1/2
  k_fill<<<(n + TB - 1) / TB, TB, 0, stream>>>(dis, 1.0f, n);
  k_count_deg<<<(e + TB - 1) / TB, TB, 0, stream>>>(dis, colIdx, e);
  k_rsqrt<<<(n + TB - 1) / TB, TB, 0, stream>>>(dis, n);

  auto gcn = [&](const float* X, const float* Wm, const float* bias,
                 int K, int Hout, float* dst, int ldOut, int colOff) {
    int mt = (n + 15) / 16;
    dim3 grid((mt + 3) / 4);
    if (Hout == 32) {
      k_gemm_wmma<2><<<grid, 128, 0, stream>>>(X, Wm, T, n, K);
    } else {
      k_gemm_wmma<8><<<grid, 128, 0, stream>>>(X, Wm, T, n, K);
    }
    long long nw = (long long)n * Hout;
    k_init_out<<<(int)((nw + TB - 1) / TB), TB, 0, stream>>>(dst, T, dis, bias, n, Hout,
                                                             ldOut, colOff);
    long long tot = (long long)e * (Hout >> 2);
    k_scatter<<<(int)((tot + TB - 1) / TB), TB, 0, stream>>>(dst, T, dis, rowIdx, colIdx, e,
                                                             Hout, ldOut, colOff);
  };
  auto tanhip = [&](float* dst, int width, int ldOut, int colOff) {
    long long nw = (long long)n * width;
    k_tanh_ip<<<(int)((nw + TB - 1) / TB), TB, 0, stream>>>(dst, n, width, ldOut, colOff);
  };

  // encoder
  gcn(feature, W_f2h, b_f2h, F, H, h, 2 * H, 0);        tanhip(h, H, 2 * H, 0);
  gcn(condition, W_c2h_e, b_c2h_e, C, H, h, 2 * H, H);  tanhip(h, H, 2 * H, H);
  gcn(h, W_mean, b_mean, 2 * H, L, meanOut, L, 0);
  gcn(h, W_logvar, b_logvar, 2 * H, L, lvOut, L, 0);
  k_z<<<(n * L + TB - 1) / TB, TB, 0, stream>>>(zOut, meanOut, lvOut, noise, n * L);
  // decoder
  gcn(zOut, W_z2h, b_z2h, L, H, hd, 2 * H, 0);          tanhip(hd, H, 2 * H, 0);
  gcn(condition, W_c2h_d, b_c2h_d, C, H, hd, 2 * H, H); tanhip(hd, H, 2 * H, H);
  gcn(hd, W_out, b_out, 2 * H, F, outOut, F, 0);
}